// GATNet_46119358824965
// MI455X (gfx1250) — compile-verified
//
#include <hip/hip_runtime.h>
#include <math.h>

typedef __attribute__((ext_vector_type(2))) float v2f;
typedef __attribute__((ext_vector_type(8))) float v8f;

// ---------------------------------------------------------------------------
// WMMA GEMM:  Y[M x OC] = X[M x K] * W[OC x K]^T  (+ bias[OC] if non-null)
// One wave per 16x16 output tile; K a template constant so the whole K-loop
// fully unrolls into a chain of V_WMMA_F32_16X16X4_F32 (32 for K=128).
// Requires M % 16 == 0 (50000 = 3125*16), OC % 16 == 0, K % 4 == 0.
// ---------------------------------------------------------------------------
template <int K>
__global__ void gemm_xwt_wmma(const float* __restrict__ X,
                              const float* __restrict__ W,
                              const float* __restrict__ bias,
                              float* __restrict__ Y,
                              int tilesM, int OC) {
  const int tileM = blockIdx.x * (blockDim.x >> 5) + (threadIdx.x >> 5);
  if (tileM >= tilesM) return;                 // wave-uniform: EXEC stays all-1s
  const int tileN = blockIdx.y;
  const int lane  = threadIdx.x & 31;
  const int r16   = lane & 15;                 // A: row (M), B: col (N), D: col (N)
  const int kh    = lane >> 4;                 // K-half selector (0 or 1)

  // A[m][k]: lane holds X[tileM*16 + r16][k0 + 2*kh + {0,1}]
  // B[k][n] = W[n][k]: lane holds W[tileN*16 + r16][k0 + 2*kh + {0,1}]
  const float* px = X + (size_t)(tileM * 16 + r16) * K + 2 * kh;
  const float* pw = W + (size_t)(tileN * 16 + r16) * K + 2 * kh;

  v8f acc = {};
#pragma unroll
  for (int k0 = 0; k0 < K; k0 += 4) {
    v2f a = *(const v2f*)(px + k0);
    v2f b = *(const v2f*)(pw + k0);
    acc = __builtin_amdgcn_wmma_f32_16x16x4_f32(
        /*neg_a=*/false, a, /*neg_b=*/false, b,
        /*c_mod=*/(short)0, acc, /*reuse_a=*/false, /*reuse_b=*/false);
  }

  const float bv = bias ? bias[tileN * 16 + r16] : 0.0f;
  // D layout: acc[j] = D[j + 8*kh][r16]
  float* py = Y + (size_t)(tileM * 16 + 8 * kh) * OC + tileN * 16 + r16;
#pragma unroll
  for (int j = 0; j < 8; ++j) py[(size_t)j * OC] = acc[j] + bv;
}

// ---------------------------------------------------------------------------
// Utility kernels
// ---------------------------------------------------------------------------
__global__ void fill_kernel(float* __restrict__ p, float v, long long n) {
  long long i = (long long)blockIdx.x * blockDim.x + threadIdx.x;
  if (i < n) p[i] = v;
}

__global__ void init_bias_kernel(float* __restrict__ h, const float* __restrict__ bias,
                                 long long total, int C) {
  long long i = (long long)blockIdx.x * blockDim.x + threadIdx.x;
  if (i < total) h[i] = bias[(int)(i % C)];
}

__device__ __forceinline__ void atomicMaxF(float* addr, float v) {
  if (v >= 0.0f) atomicMax((int*)addr, __float_as_int(v));
  else           atomicMin((unsigned int*)addr, __float_as_uint(v));
}

__device__ __forceinline__ void edge_sd(const int* __restrict__ ei, long long E,
                                        long long e, int& s, int& d) {
  if (e < E) { s = ei[e]; d = ei[E + e]; }
  else       { s = d = (int)(e - E); }        // self-loops appended
}

// ---------------------------------------------------------------------------
// GATv2 edge phase (H heads of C channels, HC = H*C)
// ---------------------------------------------------------------------------
template <int H, int C>
__global__ void edge_logits(const float* __restrict__ xl, const float* __restrict__ xr,
                            const float* __restrict__ att, const int* __restrict__ ei,
                            long long E, long long ET,
                            float* __restrict__ logits, float* __restrict__ lmax) {
  long long idx = (long long)blockIdx.x * blockDim.x + threadIdx.x;
  if (idx >= ET * H) return;
  long long e = idx / H;
  int h = (int)(idx % H);
  int s, d; edge_sd(ei, E, e, s, d);
  const float* pl = xl + (size_t)s * (H * C) + h * C;
  const float* pr = xr + (size_t)d * (H * C) + h * C;
  float acc = 0.0f;
#pragma unroll
  for (int c = 0; c < C; ++c) {
    float m = pl[c] + pr[c];
    m = (m > 0.0f) ? m : 0.2f * m;            // LeakyReLU(0.2)
    acc += att[h * C + c] * m;
  }
  logits[idx] = acc;
  atomicMaxF(&lmax[(size_t)d * H + h], acc);
}

template <int H>
__global__ void edge_expsum(float* __restrict__ la, const float* __restrict__ lmax,
                            float* __restrict__ denom, const int* __restrict__ ei,
                            long long E, long long ET) {
  long long idx = (long long)blockIdx.x * blockDim.x + threadIdx.x;
  if (idx >= ET * H) return;
  long long e = idx / H;
  int h = (int)(idx % H);
  int s, d; edge_sd(ei, E, e, s, d);
  float a = expf(la[idx] - lmax[(size_t)d * H + h]);
  la[idx] = a;                                 // overwrite logits with exp()
  atomicAdd(&denom[(size_t)d * H + h], a);
}

// One thread per (edge, float4 channel group): b128 gather + 4 atomic adds.
template <int H, int C>
__global__ void edge_scatter(const float* __restrict__ xl, const float* __restrict__ a,
                             const float* __restrict__ denom, const int* __restrict__ ei,
                             long long E, long long ET, float* __restrict__ out) {
  const int HC = H * C;
  const int G  = HC / 4;                       // float4 groups per edge
  long long idx = (long long)blockIdx.x * blockDim.x + threadIdx.x;
  if (idx >= ET * G) return;
  long long e = idx / G;
  int g  = (int)(idx % G);
  int cc = g * 4;
  int h  = cc / C;                             // all 4 channels share one head (C%4==0)
  int s, d; edge_sd(ei, E, e, s, d);
  float alpha = a[e * H + h] / (denom[(size_t)d * H + h] + 1e-16f);
  const float4 v = *(const float4*)(xl + (size_t)s * HC + cc);
  float* po = out + (size_t)d * HC + cc;
  atomicAdd(po + 0, v.x * alpha);
  atomicAdd(po + 1, v.y * alpha);
  atomicAdd(po + 2, v.z * alpha);
  atomicAdd(po + 3, v.w * alpha);
}

// ---------------------------------------------------------------------------
// BatchNorm (biased variance, eps=1e-5) + ELU (+ optional residual)
// ---------------------------------------------------------------------------
__global__ void bn_stats(const float* __restrict__ h, float* __restrict__ mu,
                         float* __restrict__ rsig, int N, int C) {
  __shared__ float ss[256];
  __shared__ float sq[256];
  int c = blockIdx.x;
  float s = 0.0f, q = 0.0f;
  for (int n = threadIdx.x; n < N; n += blockDim.x) {
    float v = h[(size_t)n * C + c];
    s += v; q += v * v;
  }
  ss[threadIdx.x] = s; sq[threadIdx.x] = q;
  __syncthreads();
  for (int st = 128; st > 0; st >>= 1) {
    if ((int)threadIdx.x < st) {
      ss[threadIdx.x] += ss[threadIdx.x + st];
      sq[threadIdx.x] += sq[threadIdx.x + st];
    }
    __syncthreads();
  }
  if (threadIdx.x == 0) {
    float m   = ss[0] / (float)N;
    float var = sq[0] / (float)N - m * m;
    mu[c]   = m;
    rsig[c] = rsqrtf(var + 1e-5f);
  }
}

__global__ void bn_apply_elu(float* __restrict__ h, const float* __restrict__ mu,
                             const float* __restrict__ rsig, const float* __restrict__ g,
                             const float* __restrict__ b, const float* __restrict__ resid,
                             long long total, int C) {
  long long i = (long long)blockIdx.x * blockDim.x + threadIdx.x;
  if (i >= total) return;
  int c = (int)(i % C);
  float v = (h[i] - mu[c]) * rsig[c] * g[c] + b[c];
  if (resid) v += resid[i];
  h[i] = (v > 0.0f) ? v : expm1f(v);           // ELU
}

// ---------------------------------------------------------------------------
// Output head: out[n,:2] = h2[n,:32] @ lin_W^T + lin_b
// ---------------------------------------------------------------------------
__global__ void final_linear(const float* __restrict__ h2, const float* __restrict__ W,
                             const float* __restrict__ b, float* __restrict__ out, int N) {
  int n = blockIdx.x * blockDim.x + threadIdx.x;
  if (n >= N) return;
  float a0 = b[0], a1 = b[1];
  const float* p = h2 + (size_t)n * 32;
#pragma unroll
  for (int c = 0; c < 32; ++c) {
    float v = p[c];
    a0 += v * W[c];
    a1 += v * W[32 + c];
  }
  out[(size_t)n * 2 + 0] = a0;
  out[(size_t)n * 2 + 1] = a1;
}

// ---------------------------------------------------------------------------
extern "C" void kernel_launch(void* const* d_in, const int* in_sizes, int n_in,
                              void* d_out, int out_size, void* d_ws, size_t ws_size,
                              hipStream_t stream) {
  const float* x     = (const float*)d_in[0];
  const int*   ei    = (const int*)  d_in[1];
  const float* Wl1   = (const float*)d_in[2];
  const float* bl1   = (const float*)d_in[3];
  const float* Wr1   = (const float*)d_in[4];
  const float* br1   = (const float*)d_in[5];
  const float* att1  = (const float*)d_in[6];
  const float* bias1 = (const float*)d_in[7];
  const float* Wl2   = (const float*)d_in[8];
  const float* bl2   = (const float*)d_in[9];
  const float* Wr2   = (const float*)d_in[10];
  const float* br2   = (const float*)d_in[11];
  const float* att2  = (const float*)d_in[12];
  const float* bias2 = (const float*)d_in[13];
  const float* bn1g  = (const float*)d_in[14];
  const float* bn1b  = (const float*)d_in[15];
  const float* bn2g  = (const float*)d_in[16];
  const float* bn2b  = (const float*)d_in[17];
  const float* skW   = (const float*)d_in[18];
  const float* linW  = (const float*)d_in[19];
  const float* linb  = (const float*)d_in[20];

  const int IN = 128, HC = 128, HID = 32, H1 = 4;
  const long long N  = in_sizes[0] / IN;      // 50000
  const long long E  = in_sizes[1] / 2;       // 800000
  const long long ET = E + N;                 // + self loops

  // --- workspace layout (floats) ---
  float* ws    = (float*)d_ws;
  float* xl1   = ws;                          // N*HC
  float* xr1   = xl1 + N * HC;                // N*HC
  float* h1    = xr1 + N * HC;                // N*HC
  float* ebuf  = h1 + N * HC;                 // ET*H1 (logits -> exp, both layers)
  float* lmax  = ebuf + ET * H1;              // N*H1
  float* den   = lmax + N * H1;               // N*H1
  float* resid = den + N * H1;                // N*HID
  float* mu    = resid + N * HID;             // 128
  float* rsig  = mu + 128;                    // 128
  // layer-2 buffers alias the (freed) xl1 region: 4*N*HID == N*HC
  float* xl2 = xl1;
  float* xr2 = xl1 + N * HID;
  float* h2  = xl1 + 2 * (N * HID);

  const int tilesM = (int)((N + 15) / 16);    // 3125, exact
  const dim3 blk(256);                        // 8 waves/block
  const unsigned gmx = (unsigned)((tilesM + 7) / 8);

  // ---- residual = x @ skip_W^T ----
  gemm_xwt_wmma<128><<<dim3(gmx, HID / 16), blk, 0, stream>>>(x, skW, nullptr, resid, tilesM, HID);
  // ---- conv1 node transforms ----
  gemm_xwt_wmma<128><<<dim3(gmx, HC / 16), blk, 0, stream>>>(x, Wl1, bl1, xl1, tilesM, HC);
  gemm_xwt_wmma<128><<<dim3(gmx, HC / 16), blk, 0, stream>>>(x, Wr1, br1, xr1, tilesM, HC);

  // ---- conv1 edge phase ----
  long long nH1 = N * H1;
  fill_kernel<<<(unsigned)((nH1 + 255) / 256), blk, 0, stream>>>(lmax, -INFINITY, nH1);
  fill_kernel<<<(unsigned)((nH1 + 255) / 256), blk, 0, stream>>>(den, 0.0f, nH1);
  init_bias_kernel<<<(unsigned)((N * HC + 255) / 256), blk, 0, stream>>>(h1, bias1, N * HC, HC);
  long long t1 = ET * H1;
  edge_logits<4, 32><<<(unsigned)((t1 + 255) / 256), blk, 0, stream>>>(xl1, xr1, att1, ei, E, ET, ebuf, lmax);
  edge_expsum<4><<<(unsigned)((t1 + 255) / 256), blk, 0, stream>>>(ebuf, lmax, den, ei, E, ET);
  long long t1s = ET * (HC / 4);
  edge_scatter<4, 32><<<(unsigned)((t1s + 255) / 256), blk, 0, stream>>>(xl1, ebuf, den, ei, E, ET, h1);

  // ---- bn1 + elu ----
  bn_stats<<<HC, blk, 0, stream>>>(h1, mu, rsig, (int)N, HC);
  bn_apply_elu<<<(unsigned)((N * HC + 255) / 256), blk, 0, stream>>>(h1, mu, rsig, bn1g, bn1b, nullptr, N * HC, HC);

  // ---- conv2 node transforms ----
  gemm_xwt_wmma<128><<<dim3(gmx, HID / 16), blk, 0, stream>>>(h1, Wl2, bl2, xl2, tilesM, HID);
  gemm_xwt_wmma<128><<<dim3(gmx, HID / 16), blk, 0, stream>>>(h1, Wr2, br2, xr2, tilesM, HID);

  // ---- conv2 edge phase (H=1) ----
  fill_kernel<<<(unsigned)((N + 255) / 256), blk, 0, stream>>>(lmax, -INFINITY, N);
  fill_kernel<<<(unsigned)((N + 255) / 256), blk, 0, stream>>>(den, 0.0f, N);
  init_bias_kernel<<<(unsigned)((N * HID + 255) / 256), blk, 0, stream>>>(h2, bias2, N * HID, HID);
  edge_logits<1, 32><<<(unsigned)((ET + 255) / 256), blk, 0, stream>>>(xl2, xr2, att2, ei, E, ET, ebuf, lmax);
  edge_expsum<1><<<(unsigned)((ET + 255) / 256), blk, 0, stream>>>(ebuf, lmax, den, ei, E, ET);
  long long t2s = ET * (HID / 4);
  edge_scatter<1, 32><<<(unsigned)((t2s + 255) / 256), blk, 0, stream>>>(xl2, ebuf, den, ei, E, ET, h2);

  // ---- bn2 + residual + elu ----
  bn_stats<<<HID, blk, 0, stream>>>(h2, mu, rsig, (int)N, HID);
  bn_apply_elu<<<(unsigned)((N * HID + 255) / 256), blk, 0, stream>>>(h2, mu, rsig, bn2g, bn2b, resid, N * HID, HID);

  // ---- output head ----
  final_linear<<<(unsigned)((N + 255) / 256), blk, 0, stream>>>(h2, linW, linb, (float*)d_out, (int)N);
}